// PatchSim_11802570130030
// MI455X (gfx1250) — compile-verified
//
#include <hip/hip_runtime.h>

typedef __attribute__((ext_vector_type(2))) float v2f;
typedef __attribute__((ext_vector_type(8))) float v8f;

#define PW 32
#define PH 32
#define C_DIM 128
#define WDIM 256
#define PTOT (PW * PH)

__device__ __forceinline__ float clip1(float v) {
    return __builtin_fminf(__builtin_fmaxf(v, -1.0f), 1.0f);
}

__global__ __launch_bounds__(256) void PatchSim_wmma_kernel(
    const float* __restrict__ feat,
    const int*   __restrict__ patch_ids,
    float*       __restrict__ out)
{
    __shared__ float q_s[C_DIM];

    const int bn = blockIdx.x;          // 0 .. B*N-1 (1024)
    const int b  = bn >> 8;             // / 256
    const int n  = bn & 255;            // % 256

    const int qid   = patch_ids[n];     // 0 .. 65535
    const int pos_x = qid >> 8;         // qid / 256
    const int pos_y = qid & 255;        // qid % 256

    int sx = pos_x - PW / 2;
    if (sx < 0) sx = 0;
    if (sx > WDIM - PW) sx = WDIM - PW;
    int sy = pos_y - PH / 2;
    if (sy < 0) sy = 0;
    if (sy > WDIM - PH) sy = WDIM - PH;

    const size_t plane = (size_t)WDIM * WDIM;               // 65536
    const float* featb = feat + (size_t)b * C_DIM * plane;

    // Load + clip the query vector (128 floats) into LDS once per block.
    const int t = threadIdx.x;
    if (t < C_DIM) {
        q_s[t] = clip1(featb[(size_t)t * plane + (size_t)pos_x * WDIM + pos_y]);
    }
    __syncthreads();

    const int lane  = t & 31;
    const int wave  = t >> 5;
    const int i     = lane & 15;            // M index (p within 16-tile)
    const int halfk = (lane >> 4) << 1;     // K offset: 0 for lanes 0-15, 2 for 16-31

    float* outp = out + (size_t)bn * PTOT;

    // 64 p-tiles of 16; 8 waves -> 8 tiles per wave.
    for (int j = 0; j < 8; ++j) {
        const int p0 = (wave * 8 + j) * 16;
        const int p  = p0 + i;
        const int x  = sx + (p >> 5);       // p / PH
        const int y  = sy + (p & 31);       // p % PH
        const float* kptr = featb + (size_t)halfk * plane + (size_t)x * WDIM + y;

        v8f acc = {};
        #pragma unroll 4
        for (int kc = 0; kc < 32; ++kc) {
            v2f a, bb;
            // A: key-window values kw[c, p] -- K in {halfk, halfk+1} of this chunk.
            a.x = clip1(kptr[0]);
            a.y = clip1(kptr[plane]);
            // B: query values broadcast across all 16 N columns (same LDS addr
            // per half-wave -> LDS broadcast read).
            const int c0 = kc * 4 + halfk;
            bb.x = q_s[c0];
            bb.y = q_s[c0 + 1];
            // D = A(16x4) * B(4x16) + C, f32: v_wmma_f32_16x16x4_f32
            acc = __builtin_amdgcn_wmma_f32_16x16x4_f32(
                false, a, false, bb, (short)0, acc, false, false);
            kptr += 4 * plane;              // advance 4 channel planes
        }

        // Every D column is identical; lane 0 holds M=0..7 in v0..7, lane 16
        // holds M=8..15. Each of the two lanes stores 8 contiguous floats.
        if ((lane & 15) == 0) {
            float* o = outp + p0 + ((lane >> 4) << 3);
            #pragma unroll
            for (int k2 = 0; k2 < 8; ++k2) o[k2] = acc[k2];
        }
    }
}

extern "C" void kernel_launch(void* const* d_in, const int* in_sizes, int n_in,
                              void* d_out, int out_size, void* d_ws, size_t ws_size,
                              hipStream_t stream) {
    (void)in_sizes; (void)n_in; (void)out_size; (void)d_ws; (void)ws_size;
    const float* feat      = (const float*)d_in[0];
    const int*   patch_ids = (const int*)d_in[1];
    float*       out       = (float*)d_out;

    dim3 grid(4 * 256);   // B * N blocks, one (b, n) pair each
    dim3 block(256);      // 8 wave32 waves
    hipLaunchKernelGGL(PatchSim_wmma_kernel, grid, block, 0, stream,
                       feat, patch_ids, out);
}